// PointerNet_76871324664558
// MI455X (gfx1250) — compile-verified
//
#include <hip/hip_runtime.h>
#include <hip/hip_bf16.h>

typedef __attribute__((ext_vector_type(2))) float v2f;
typedef __attribute__((ext_vector_type(8))) float v8f;

#define L_Q   512
#define L_A   512
#define NB    4
#define QS    256
#define DSZ   512
#define NH    128
#define ATILE 8

// ---------------------------------------------------------------------------
// Kernel 1: Y = X @ W + bias  via V_WMMA_F32_16X16X4_F32  (full fp32 precision)
// X: (M, K) row-major, W: (K, NH) row-major, Y: (M, NH).
// One wave per 16x16 output tile, 8 waves per block. K stepped by 4 per WMMA.
// ---------------------------------------------------------------------------
__global__ __launch_bounds__(256) void proj_wmma_kernel(
    const float* __restrict__ X,
    const float* __restrict__ W,
    const float* __restrict__ bias,
    float* __restrict__ Y,
    int M, int K)
{
  const int lane   = threadIdx.x & 31;
  const int wave   = threadIdx.x >> 5;
  const int tilesN = NH / 16;                 // 8
  const int tile   = blockIdx.x * 8 + wave;   // global 16x16 tile id
  const int total  = (M / 16) * tilesN;
  if (tile >= total) return;                  // wave-uniform (EXEC all-ones inside)

  const int tm   = tile / tilesN;
  const int tn   = tile % tilesN;
  const int half = lane >> 4;                 // selects K-subpair {0,1} vs {2,3}
  const int l16  = lane & 15;
  const int rowA = tm * 16 + l16;             // A: M striped across lane halves
  const int colB = tn * 16 + l16;             // B/C/D: N striped across lanes

  const float* __restrict__ xrow = X + (long)rowA * K;

  v8f acc = {};
  for (int k = 0; k < K; k += 4) {
    const int ka = k + half * 2;
    // A fragment: lane holds A[rowA][ka], A[rowA][ka+1]  (8B-aligned: K even, ka even)
    v2f afrag = *(const v2f*)(xrow + ka);
    // B fragment: lane holds B[ka][colB], B[ka+1][colB]
    v2f bfrag;
    bfrag[0] = W[(long)(ka)     * NH + colB];
    bfrag[1] = W[(long)(ka + 1) * NH + colB];
    acc = __builtin_amdgcn_wmma_f32_16x16x4_f32(
        false, afrag, false, bfrag, (short)0, acc, false, false);
  }

  const float bv = bias[colB];
  #pragma unroll
  for (int r = 0; r < 8; ++r) {
    const int m = tm * 16 + r + half * 8;     // C/D layout: VGPR r -> M=r / M=r+8
    Y[(long)m * NH + colB] = acc[r] + bv;
  }
}

// ---------------------------------------------------------------------------
// Fast device math: prefer hardware V_TANH_F32 / V_EXP_F32 paths.
// ---------------------------------------------------------------------------
__device__ __forceinline__ float fast_tanh(float x) {
#if __has_builtin(__builtin_amdgcn_tanhf)
  return __builtin_amdgcn_tanhf(x);
#elif __has_builtin(__builtin_amdgcn_tanh_f32)
  return __builtin_amdgcn_tanh_f32(x);
#else
  // tanh(x) = 1 - 2/(exp(2x)+1); exp(2x) = exp2(x * 2*log2(e))
  const float e = __builtin_amdgcn_exp2f(x * 2.8853900817779268f);
  return 1.0f - 2.0f * __builtin_amdgcn_rcpf(e + 1.0f);
#endif
}

__device__ __forceinline__ float fast_exp(float x) {
  return __builtin_amdgcn_exp2f(x * 1.4426950408889634f);  // v_exp_f32 path
}

// ---------------------------------------------------------------------------
// Kernel 2: fused scores + masked softmax.
// Block = 256 threads handles (b, 8 consecutive a). Thread owns q = tid, tid+256.
// scores[a,q,b] = b2 + sum_h tanh(qt[q,b,h] + ht[a,b,h]) * w2[h]
// softmax over q (max over ALL q, mask applied AFTER exp — faithful to ref).
// out: (L_A, NB, L_Q).
// ---------------------------------------------------------------------------
__global__ __launch_bounds__(256) void attn_scores_softmax_kernel(
    const float* __restrict__ qt,            // (L_Q, NB, NH)
    const float* __restrict__ ht,            // (L_A, NB, NH)
    const unsigned char* __restrict__ qmask, // (L_Q, NB) bool
    const float* __restrict__ w2,            // (NH)
    const float* __restrict__ b2,            // (1)
    float* __restrict__ out)                 // (L_A, NB, L_Q)
{
  __shared__ float s_ht[ATILE][NH];
  __shared__ float s_w2[NH];
  __shared__ float s_red[ATILE][8];

  const int tid = threadIdx.x;
  const int b   = blockIdx.x & (NB - 1);
  const int a0  = (blockIdx.x >> 2) * ATILE;

  for (int idx = tid; idx < ATILE * NH; idx += 256) {
    const int i = idx >> 7, h = idx & (NH - 1);
    s_ht[i][h] = ht[((long)(a0 + i) * NB + b) * NH + h];
  }
  if (tid < NH) s_w2[tid] = w2[tid];
  __syncthreads();

  const int q0 = tid, q1 = tid + 256;
  const float* __restrict__ qp0 = qt + ((long)q0 * NB + b) * NH;
  const float* __restrict__ qp1 = qt + ((long)q1 * NB + b) * NH;

  float acc0[ATILE], acc1[ATILE];
  #pragma unroll
  for (int i = 0; i < ATILE; ++i) { acc0[i] = 0.f; acc1[i] = 0.f; }

  // 16 tanh + 16 FMA per h-iteration; qt element reused across 8 a's (registers).
  for (int h = 0; h < NH; ++h) {
    const float w  = s_w2[h];
    const float x0 = qp0[h];
    const float x1 = qp1[h];
    #pragma unroll
    for (int i = 0; i < ATILE; ++i) {
      const float hv = s_ht[i][h];
      acc0[i] += w * fast_tanh(x0 + hv);
      acc1[i] += w * fast_tanh(x1 + hv);
    }
  }

  const float b2v = b2[0];
  #pragma unroll
  for (int i = 0; i < ATILE; ++i) { acc0[i] += b2v; acc1[i] += b2v; }

  const int lane = tid & 31, wid = tid >> 5;

  // ---- block max over q (512 values) for each of the 8 a's ----
  float red[ATILE];
  #pragma unroll
  for (int i = 0; i < ATILE; ++i) red[i] = fmaxf(acc0[i], acc1[i]);
  #pragma unroll
  for (int off = 16; off >= 1; off >>= 1) {
    #pragma unroll
    for (int i = 0; i < ATILE; ++i)
      red[i] = fmaxf(red[i], __shfl_xor(red[i], off, 32));
  }
  if (lane == 0) {
    #pragma unroll
    for (int i = 0; i < ATILE; ++i) s_red[i][wid] = red[i];
  }
  __syncthreads();
  float mx[ATILE];
  #pragma unroll
  for (int i = 0; i < ATILE; ++i) {
    float m = s_red[i][0];
    #pragma unroll
    for (int w = 1; w < 8; ++w) m = fmaxf(m, s_red[i][w]);
    mx[i] = m;
  }
  __syncthreads();  // before s_red reuse

  // ---- exp, then mask (AFTER exp, as in reference) ----
  const float m0 = (float)qmask[(long)q0 * NB + b];
  const float m1 = (float)qmask[(long)q1 * NB + b];
  float e0[ATILE], e1[ATILE];
  #pragma unroll
  for (int i = 0; i < ATILE; ++i) {
    e0[i] = fast_exp(acc0[i] - mx[i]) * m0;
    e1[i] = fast_exp(acc1[i] - mx[i]) * m1;
  }

  // ---- block sum over q ----
  #pragma unroll
  for (int i = 0; i < ATILE; ++i) red[i] = e0[i] + e1[i];
  #pragma unroll
  for (int off = 16; off >= 1; off >>= 1) {
    #pragma unroll
    for (int i = 0; i < ATILE; ++i)
      red[i] += __shfl_xor(red[i], off, 32);
  }
  if (lane == 0) {
    #pragma unroll
    for (int i = 0; i < ATILE; ++i) s_red[i][wid] = red[i];
  }
  __syncthreads();
  #pragma unroll
  for (int i = 0; i < ATILE; ++i) {
    float s = s_red[i][0];
    #pragma unroll
    for (int w = 1; w < 8; ++w) s += s_red[i][w];
    const float inv = __builtin_amdgcn_rcpf(s);
    float* orow = out + ((long)(a0 + i) * NB + b) * L_Q;
    orow[q0] = e0[i] * inv;   // consecutive tid -> consecutive q: coalesced
    orow[q1] = e1[i] * inv;
  }
}

// ---------------------------------------------------------------------------
extern "C" void kernel_launch(void* const* d_in, const int* in_sizes, int n_in,
                              void* d_out, int out_size, void* d_ws, size_t ws_size,
                              hipStream_t stream) {
  (void)in_sizes; (void)n_in; (void)out_size; (void)ws_size;

  const float*         query   = (const float*)d_in[0];         // (L_Q, NB, QS)
  const float*         dstates = (const float*)d_in[1];         // (L_A, NB, DSZ)
  const unsigned char* qmask   = (const unsigned char*)d_in[2]; // (L_Q, NB) bool
  const float*         Wq      = (const float*)d_in[3];         // (QS, NH)
  const float*         bq      = (const float*)d_in[4];         // (NH)
  const float*         Wh      = (const float*)d_in[5];         // (DSZ, NH)
  const float*         bh      = (const float*)d_in[6];         // (NH)
  const float*         w2      = (const float*)d_in[7];         // (NH)
  const float*         b2      = (const float*)d_in[8];         // (1)
  float*               out     = (float*)d_out;                 // (L_A, NB, L_Q)

  // workspace: qt (1 MB) | ht (1 MB)
  float* qt = (float*)d_ws;
  float* ht = qt + (size_t)L_Q * NB * NH;

  // Projections: M = 2048 rows (q*B+b flattened), 16x16 tiles, 8 waves/block.
  const int tilesQ = (L_Q * NB / 16) * (NH / 16);  // 1024 tiles -> 128 blocks
  proj_wmma_kernel<<<tilesQ / 8, 256, 0, stream>>>(query, Wq, bq, qt, L_Q * NB, QS);
  const int tilesH = (L_A * NB / 16) * (NH / 16);  // 1024 tiles -> 128 blocks
  proj_wmma_kernel<<<tilesH / 8, 256, 0, stream>>>(dstates, Wh, bh, ht, L_A * NB, DSZ);

  // Fused scores + softmax: one block per (8 a's, b) -> 256 blocks.
  attn_scores_softmax_kernel<<<(L_A / ATILE) * NB, 256, 0, stream>>>(
      qt, ht, qmask, w2, b2, out);
}